// resnet_with_att_conv_regions_57990648430675
// MI455X (gfx1250) — compile-verified
//
#include <hip/hip_runtime.h>

// CDNA5 / gfx1250, wave32.  Single fused kernel:
//   fc_feats  [B, C]      = spatial mean            (row 36 of mask GEMM)
//   att_feats [B, HW, C]  = transpose of images     (from the same LDS tile)
//   bu_feats  [B, R, C]   = per-box average pool    (mask-GEMM, V_WMMA_F32_16X16X4_F32)
//
// Roofline: images read once (205 MB) + outputs written once (244 MB)
// => ~450 MB @ 23.3 TB/s ~= 19 us.  The 48x196 mask matrix is built in LDS
// (division-free), so the WMMA K-loop is pure ds_load_b64 + v_wmma.

typedef float v2f __attribute__((ext_vector_type(2)));
typedef float v8f __attribute__((ext_vector_type(8)));

constexpr int B_   = 128;
constexpr int C_   = 2048;
constexpr int P_   = 14;
constexpr int HW_  = 196;    // 14*14 = 49 float4s
constexpr int R_   = 36;
constexpr int XPIT = 198;    // image-tile pitch: even (8B-aligned b64 frags),
                             // 6*ch mod 64 injective over 32 lanes -> frag reads conflict-free

__global__ __launch_bounds__(384) void fused_region_kernel(
    const float* __restrict__ img, const float* __restrict__ boxes,
    float* __restrict__ fc, float* __restrict__ att, float* __restrict__ bu) {
  __shared__ __align__(16) float xt[64 * XPIT];    // 64 channels x 196 pixels
  __shared__ __align__(16) float mask[48 * HW_];   // 48 mask rows x 196 pixels
  __shared__ int   bxi[R_][4];
  __shared__ float binv[R_];

  const int b    = blockIdx.y;
  const int c0   = blockIdx.x * 64;
  const int t    = threadIdx.x;
  const int wave = t >> 5;
  const int lane = t & 31;

  // --- box setup: replicate jnp.round (rintf = half-to-even) + degenerate fix
  if (t < R_) {
    const float* bp = boxes + ((size_t)b * R_ + t) * 4;
    int x1 = (int)rintf(bp[0] * (float)P_);
    int y1 = (int)rintf(bp[1] * (float)P_);
    int x2 = (int)rintf(bp[2] * (float)P_);
    int y2 = (int)rintf(bp[3] * (float)P_);
    const bool eqx = (x1 == x2);
    if (eqx && x2 < P_) x2 += 1;
    if (eqx && x2 >= P_ && x1 > 0 && x1 == x2) x1 -= 1;
    const bool eqy = (y1 == y2);
    if (eqy && y2 < P_) y2 += 1;
    if (eqy && y2 >= P_ && y1 > 0 && y1 == y2) y1 -= 1;
    bxi[t][0] = x1; bxi[t][1] = y1; bxi[t][2] = x2; bxi[t][3] = y2;
    binv[t] = 1.0f / (float)((y2 - y1) * (x2 - x1));
  }

  // --- stage 64 x 196 image tile (float4-coalesced loads)
  const float* src = img + ((size_t)b * C_ + c0) * HW_;
  for (int i = t; i < 64 * 49; i += 384) {
    const int ch = i / 49;
    const int v  = i % 49;
    const float4 d = *(const float4*)(src + (size_t)ch * HW_ + 4 * v);
    float* row = &xt[ch * XPIT + 4 * v];
    row[0] = d.x; row[1] = d.y; row[2] = d.z; row[3] = d.w;
  }
  __syncthreads();   // bxi/binv ready

  // --- build 48x196 mask, division-free: wave owns 4 rows, lanes sweep
  //     hw = lane + 32j with incremental (px,py): hw+=32 => px+=4, py+=2 (+carry)
  {
    int px0 = lane, py0 = 0;
    if (px0 >= P_) { px0 -= P_; py0 = 1; }
    if (px0 >= P_) { px0 -= P_; py0 = 2; }   // lane <= 31 -> py0 <= 2
    for (int r = 0; r < 4; ++r) {
      const int row = wave * 4 + r;
      int x1 = 0, y1 = 0, x2 = 0, y2 = 0;
      float w = 0.0f;
      if (row < R_) {
        x1 = bxi[row][0]; y1 = bxi[row][1];
        x2 = bxi[row][2]; y2 = bxi[row][3];
        w  = binv[row];
      } else if (row == R_) {                 // fc row: global mean
        x2 = P_; y2 = P_; w = 1.0f / (float)HW_;
      }                                        // rows 37..47 stay 0
      int hw = lane, px = px0, py = py0;
      #pragma unroll
      for (int j = 0; j < 7; ++j) {
        if (hw < HW_) {
          mask[row * HW_ + hw] =
              (px >= x1 && px < x2 && py >= y1 && py < y2) ? w : 0.0f;
        }
        hw += 32; px += 4; py += 2;
        if (px >= P_) { px -= P_; py += 1; }
      }
    }
  }
  __syncthreads();   // xt + mask ready

  // --- att_feats transpose: float4 stores (lane gathers 4 consecutive channels)
  {
    const int cl4 = (t & 15) * 4;            // channel group 0,4,...,60
    float* dst0 = att + (size_t)b * HW_ * C_ + c0 + cl4;
    for (int hw = t >> 4; hw < HW_; hw += 24) {
      float4 d;
      d.x = xt[(cl4 + 0) * XPIT + hw];
      d.y = xt[(cl4 + 1) * XPIT + hw];
      d.z = xt[(cl4 + 2) * XPIT + hw];
      d.w = xt[(cl4 + 3) * XPIT + hw];
      *(float4*)(dst0 + (size_t)hw * C_) = d;  // half-wave: contiguous 256B row
    }
  }

  // --- mask GEMM: D[48 x 64] = M[48 x 196] x X^T[196 x 64]
  // 12 waves: wave w -> m_tile = w % 3, n_tile = w / 3.
  const int m_tile = wave % 3;
  const int n_tile = wave / 3;
  const int mn     = lane & 15;   // A-row within tile / B-column within tile
  const int kh     = lane >> 4;   // K-half: +0 (lanes 0-15), +2 (lanes 16-31)

  const float* aptr = &mask[(m_tile * 16 + mn) * HW_ + 2 * kh];
  const float* bptr = &xt[(n_tile * 16 + mn) * XPIT + 2 * kh];

  v8f acc = {0.f, 0.f, 0.f, 0.f, 0.f, 0.f, 0.f, 0.f};
  for (int kk = 0; kk < 49; ++kk) {
    const v2f afrag = *(const v2f*)(aptr + 4 * kk);   // ds_load_b64, conflict-free
    const v2f bfrag = *(const v2f*)(bptr + 4 * kk);
    acc = __builtin_amdgcn_wmma_f32_16x16x4_f32(
        /*neg_a=*/false, afrag, /*neg_b=*/false, bfrag,
        /*c_mod=*/(short)0, acc, /*reuse_a=*/false, /*reuse_b=*/false);
  }

  // --- store epilogue, wave-uniform branching.
  // C/D layout: VGPR v -> row m_tile*16 + v + 8*kh, col = mn.
  const int col = c0 + n_tile * 16 + mn;
  if (m_tile < 2) {
    // rows m_tile*16 + 8*kh + v, all < 36: unconditional strided stores
    float* dst = &bu[((size_t)b * R_ + m_tile * 16 + 8 * kh) * C_ + col];
    #pragma unroll
    for (int v = 0; v < 8; ++v) dst[(size_t)v * C_] = acc[v];
  } else if (kh == 0) {
    // rows 32..35 -> bu; row 36 -> fc; rows 37..39 dropped (kh==1: 40..47 dropped)
    float* dst = &bu[((size_t)b * R_ + 32) * C_ + col];
    dst[0 * (size_t)C_] = acc[0];
    dst[1 * (size_t)C_] = acc[1];
    dst[2 * (size_t)C_] = acc[2];
    dst[3 * (size_t)C_] = acc[3];
    fc[(size_t)b * C_ + col] = acc[4];
  }
}

// ---------------------------------------------------------------------------
extern "C" void kernel_launch(void* const* d_in, const int* in_sizes, int n_in,
                              void* d_out, int out_size, void* d_ws, size_t ws_size,
                              hipStream_t stream) {
  (void)in_sizes; (void)n_in; (void)out_size; (void)d_ws; (void)ws_size;
  const float* img   = (const float*)d_in[0];
  const float* boxes = (const float*)d_in[1];

  float* out = (float*)d_out;
  float* fc  = out;                                   // [B, C]
  float* att = fc + (size_t)B_ * C_;                  // [B, HW, C]
  float* bu  = att + (size_t)B_ * HW_ * C_;           // [B, R, C]

  dim3 grid(C_ / 64, B_);
  fused_region_kernel<<<grid, 384, 0, stream>>>(img, boxes, fc, att, bu);
}